// DHT_Transform_3032246911418
// MI455X (gfx1250) — compile-verified
//
#include <hip/hip_runtime.h>

typedef __attribute__((ext_vector_type(2))) float v2f;
typedef __attribute__((ext_vector_type(4))) float v4f;
typedef __attribute__((ext_vector_type(8))) float v8f;

// out[b] = T_theta(x[b]) @ RIGHT,  RIGHT = [[1,0,0,0.2],[0,c,-s,0],[0,s,c,0.1],[0,0,0,1]]
// Computed as out^T = RIGHT^T @ T_theta^T via V_WMMA_F32_16X16X4_F32, packing 4
// batch matrices block-diagonally into the 16x16 tile so that:
//   - A (16x4)  = RIGHT^T replicated down the row blocks  -> loop-invariant
//   - B (4x16)  = four T_theta^T matrices across columns  -> only theta-dependent part
//   - D (16x16) diagonal blocks = out^T; each useful lane holds one full output
//     row in 4 consecutive VGPRs -> single coalesced global_store_b128 per lane.
__global__ void dht_wmma_kernel(const float* __restrict__ x,
                                float* __restrict__ out,
                                int n) {
  const int lane = threadIdx.x & 31;                 // wave32
  // wave id is wave-uniform: pin it to SGPRs so loop control is scalar
  const int wave = __builtin_amdgcn_readfirstlane(
      (int)((blockIdx.x * blockDim.x + threadIdx.x) >> 5));
  const int nwav = (int)((gridDim.x * blockDim.x) >> 5);

  const int  lo = lane & 15;
  const bool hi = lane >= 16;          // A/B: hi half-wave holds K=2,3
  const int  i  = lo & 3;              // row-in-block (m%4 for A, n%4 for B)

  const float c_al = 0.955336489125606f;   // cos(0.3f), RNE to f32
  const float s_al = 0.295520206661340f;   // sin(0.3f)

  // A fragment (loop-invariant): A[m][k] = RIGHT[k][m%4], lane holds K = {2*hi, 2*hi+1}
  v2f afrag;
  if (!hi) {
    afrag.x = (i == 0) ? 1.0f : ((i == 3) ? 0.2f : 0.0f);          // R[0][i]
    afrag.y = (i == 1) ? c_al : ((i == 2) ? -s_al : 0.0f);         // R[1][i]
  } else {
    afrag.x = (i == 1) ? s_al : ((i == 2) ? c_al
                               : ((i == 3) ? 0.1f : 0.0f));        // R[2][i]
    afrag.y = (i == 3) ? 1.0f : 0.0f;                              // R[3][i]
  }

  // D-fragment store mapping (lane = N + 16*(M>=8), vgpr = M%8): diagonal 4x4
  // blocks live in lanes 0-7 (blocks 0,1) and 24-31 (blocks 2,3); each useful
  // lane holds one output row in 4 consecutive VGPRs.
  const bool active    = (lane < 8) || (lane >= 24);
  const int  blk       = (lane < 8) ? (lane >> 2) : (2 + ((lane - 24) >> 2));
  const int  row       = lane & 3;
  const bool upperquad = (lane & 4) != 0;            // block rows 4-7/12-15 -> d[4..7]

  for (int b0 = wave * 4; b0 < n; b0 += nwav * 4) {
    // B fragment: B[k][n] = T_{n/4}[n%4][k]; this lane's column group = batch b0+lo/4
    int bi = b0 + (lo >> 2);
    int bc = bi < n ? bi : (n - 1);                  // clamp keeps EXEC all-ones
    float th = x[bc];
    float st, ct;
    sincosf(th, &st, &ct);                           // one shared range reduction

    v2f bfrag;
    if (!hi) {                                       // K = 0,1
      bfrag.x = (i == 0) ? ct : ((i == 1) ? st : 0.0f);   // T[i][0]
      bfrag.y = (i == 0) ? -st : ((i == 1) ? ct : 0.0f);  // T[i][1]
    } else {                                         // K = 2,3
      bfrag.x = (i == 2) ? 1.0f : 0.0f;                   // T[i][2]
      bfrag.y = (i == 3) ? 1.0f : 0.0f;                   // T[i][3]
    }

    v8f cz = {};
    // 8 args: (neg_a, A, neg_b, B, c_mod, C, reuse_a, reuse_b)
    v8f d = __builtin_amdgcn_wmma_f32_16x16x4_f32(
        false, afrag, false, bfrag, (short)0, cz, false, false);

    // Select this lane's quad with constant indices (no dynamic vector indexing)
    v4f o;
    o.x = upperquad ? d[4] : d[0];
    o.y = upperquad ? d[5] : d[1];
    o.z = upperquad ? d[6] : d[2];
    o.w = upperquad ? d[7] : d[3];

    int ob = b0 + blk;
    if (active && ob < n) {
      // out[ob][row][0..3]: 16B per lane, 256B contiguous per wave-iteration
      *(v4f*)(out + (size_t)ob * 16 + row * 4) = o;
    }
  }
}

extern "C" void kernel_launch(void* const* d_in, const int* in_sizes, int n_in,
                              void* d_out, int out_size, void* d_ws, size_t ws_size,
                              hipStream_t stream) {
  const float* x = (const float*)d_in[0];
  float* out = (float*)d_out;
  int n = in_sizes[0];                 // B (x is [B,1])

  const int threads = 256;             // 8 waves per block
  int blocks = (n + (threads / 32) * 4 - 1) / ((threads / 32) * 4);
  if (blocks > 2048) blocks = 2048;    // grid-stride loop covers the rest
  if (blocks < 1) blocks = 1;
  dht_wmma_kernel<<<blocks, threads, 0, stream>>>(x, out, n);
}